// BoxMinusMatNLLLoss_20521353740533
// MI455X (gfx1250) — compile-verified
//
#include <hip/hip_runtime.h>
#include <math.h>

typedef __attribute__((ext_vector_type(2))) float v2f;
typedef __attribute__((ext_vector_type(8))) float v8f;

#define PI_F      3.14159265358979323846f
#define HALF_PI_F 1.57079632679489661923f
#define INV_2PI_F 0.15915494309189533577f
#define EPS_F     1e-07f

// v_rcp_f32 + one Newton step (~1e-7 relative error; pivots are >= 0.5 so safe)
__device__ __forceinline__ float fast_rcp(float x) {
  float r = __builtin_amdgcn_rcpf(x);
  return r * fmaf(-x, r, 2.0f);
}

// atan2 via odd minimax poly on [0,1] + quadrant fixup (~3e-6 rad max err)
__device__ __forceinline__ float fast_atan2(float yv, float xv) {
  const float ax = fabsf(xv), ay = fabsf(yv);
  const float mx = fmaxf(ax, ay);
  const float mn = fminf(ax, ay);
  const float t  = mn * fast_rcp(mx);
  const float t2 = t * t;
  float p = -0.01172120f;
  p = fmaf(p, t2,  0.05265332f);
  p = fmaf(p, t2, -0.11643287f);
  p = fmaf(p, t2,  0.19354346f);
  p = fmaf(p, t2, -0.33262347f);
  p = fmaf(p, t2,  0.99997726f);
  float r = p * t;
  r = (ay > ax) ? (HALF_PI_F - r) : r;
  r = (xv < 0.0f) ? (PI_F - r) : r;
  return copysignf(r, yv);
}

// ---------------------------------------------------------------------------
// Wave32 + block reduction via V_WMMA_F32_16X16X4_F32 (all-ones B => row sums)
//   A row m = [p_m, 0, p_{m+16}, 0]  ->  D[m][n] = p_m + p_{m+16}
// sum(d[0..7]) per lane + shfl_xor(16) == exact f32 wave sum.
// ---------------------------------------------------------------------------
__device__ __forceinline__ float block_reduce_sum(float p, float* lds) {
#if __has_builtin(__builtin_amdgcn_wmma_f32_16x16x4_f32)
  v2f a; a[0] = p;    a[1] = 0.0f;
  v2f b; b[0] = 1.0f; b[1] = 1.0f;
  v8f c = {};
  v8f d = __builtin_amdgcn_wmma_f32_16x16x4_f32(
      false, a, false, b, (short)0, c, false, false);
  float rs = (d[0] + d[1]) + (d[2] + d[3]) + (d[4] + d[5]) + (d[6] + d[7]);
  float wsum = rs + __shfl_xor(rs, 16, 32);
#else
  float wsum = p;
  #pragma unroll
  for (int off = 16; off >= 1; off >>= 1) wsum += __shfl_xor(wsum, off, 32);
#endif
  const int lane = threadIdx.x & 31;
  const int wave = threadIdx.x >> 5;
  if (lane == 0) lds[wave] = wsum;
  __syncthreads();
  float r = 0.0f;
  if (threadIdx.x == 0) {
    const int nw = blockDim.x >> 5;
    for (int w = 0; w < nw; ++w) r += lds[w];
  }
  return r;  // valid in thread 0 only
}

// ---------------------------------------------------------------------------
// Stage 1: per-element loss, per-block partial sums (scaled by 1/N)
// ---------------------------------------------------------------------------
__global__ void __launch_bounds__(256)
bm_nll_partial(const float* __restrict__ q, const float* __restrict__ qh,
               const float* __restrict__ cov, float* __restrict__ partial,
               int n, float inv_n) {
  __shared__ float lds[8];
  const float4* q4  = (const float4*)q;
  const float4* qh4 = (const float4*)qh;
  const float4* c4  = (const float4*)cov;

  float acc = 0.0f;
  const int stride = gridDim.x * blockDim.x;
  for (int i = blockIdx.x * blockDim.x + threadIdx.x; i < n; i += stride) {
    const float4 qv = q4[i];
    const float4 hv = qh4[i];
    const float4 r0 = c4[i * 4 + 0];
    const float4 r1 = c4[i * 4 + 1];
    const float4 r2 = c4[i * 4 + 2];
    const float4 r3 = c4[i * 4 + 3];

    // box_minus: qd = [x, y, th, th]
    const float x   = qv.x - hv.x;
    const float y   = qv.y - hv.y;
    const float den = hv.w + ((hv.w == 0.0f) ? EPS_F : 0.0f);
    const float dth = qv.z - fast_atan2(hv.z, den);
    const float th  = dth - 2.0f * PI_F * floorf(fmaf(dth, INV_2PI_F, 0.5f));

    // cov is exactly symmetric SPD (A*A^T + 0.5I): use upper triangle
    //   C = [a b c d; b e f g; c f h i; d g i j]
    const float a = r0.x, b = r0.y, cc = r0.z, dd = r0.w;
    const float e = r1.y, f = r1.z, g  = r1.w;
    const float h = r2.z, ii = r2.w;
    const float j = r3.w;

    // LDL^T factorization (pivots >= 0.5 by construction)
    const float rd1 = fast_rcp(a);
    const float l21 = b  * rd1;
    const float l31 = cc * rd1;
    const float l41 = dd * rd1;
    const float d2  = fmaf(-l21, b, e);
    const float u2  = fmaf(-l31, b, f);     // pre-scaled (f - L31*b)
    const float v2  = fmaf(-l41, b, g);     // pre-scaled (g - L41*b)
    const float rd2 = fast_rcp(d2);
    const float l32 = u2 * rd2;
    const float l42 = v2 * rd2;
    const float d3  = fmaf(-l32, u2, fmaf(-l31, cc, h));
    const float w3  = fmaf(-l42, u2, fmaf(-l41, cc, ii));
    const float rd3 = fast_rcp(d3);
    const float l43 = w3 * rd3;
    const float d4  = fmaf(-l43, w3, fmaf(-l42, v2, fmaf(-l41, dd, j)));
    const float rd4 = fast_rcp(d4);

    // forward solve L z = qd, then qf = sum z_i^2 / d_i
    const float z1 = x;
    const float z2 = fmaf(-l21, z1, y);
    const float z3 = fmaf(-l32, z2, fmaf(-l31, z1, th));
    const float z4 = fmaf(-l43, z3, fmaf(-l42, z2, fmaf(-l41, z1, th)));
    const float qf = fmaf(z1 * z1, rd1,
                     fmaf(z2 * z2, rd2,
                     fmaf(z3 * z3, rd3, (z4 * z4) * rd4)));
    const float l_traj = 0.5f * qf;

    // Frobenius norm^2 via symmetry: diag^2 + 2*offdiag^2
    const float diag2 = fmaf(a, a, fmaf(e, e, fmaf(h, h, j * j)));
    const float off2  = fmaf(b, b, fmaf(cc, cc, fmaf(dd, dd,
                        fmaf(f, f, fmaf(g, g, ii * ii)))));
    const float ss    = fmaf(2.0f, off2, diag2);
    const float l_cov = 0.25f * __logf(ss);   // 0.5*log(sqrt(ss))

    acc += l_traj + l_cov;
  }

  const float r = block_reduce_sum(acc * inv_n, lds);
  if (threadIdx.x == 0) partial[blockIdx.x] = r;
}

// ---------------------------------------------------------------------------
// Stage 2: reduce block partials -> scalar
// ---------------------------------------------------------------------------
__global__ void __launch_bounds__(256)
bm_nll_finalize(const float* __restrict__ partial, float* __restrict__ out, int nb) {
  __shared__ float lds[8];
  float acc = 0.0f;
  for (int i = threadIdx.x; i < nb; i += blockDim.x) acc += partial[i];
  const float r = block_reduce_sum(acc, lds);
  if (threadIdx.x == 0) out[0] = r;
}

extern "C" void kernel_launch(void* const* d_in, const int* in_sizes, int n_in,
                              void* d_out, int out_size, void* d_ws, size_t ws_size,
                              hipStream_t stream) {
  const float* q   = (const float*)d_in[0];
  const float* qh  = (const float*)d_in[1];
  const float* cov = (const float*)d_in[2];
  // d_in[3] = device index (unused)

  const int n = in_sizes[0] / 4;  // B*T elements
  const int threads = 256;
  const int blocks = 2048;        // 8 KB of partials in d_ws, 2 elem/thread
  float* partial = (float*)d_ws;

  bm_nll_partial<<<blocks, threads, 0, stream>>>(q, qh, cov, partial, n,
                                                 1.0f / (float)n);
  bm_nll_finalize<<<1, threads, 0, stream>>>(partial, (float*)d_out, blocks);
}